// RUNG_35991825941090
// MI455X (gfx1250) — compile-verified
//
#include <hip/hip_runtime.h>
#include <hip/hip_bf16.h>

// ---------------------------------------------------------------------------
// MCP-IRLS graph propagation for MI455X (gfx1250).
// Flash-attention-style fusion: Gram tile (f32 WMMA) -> MCP weight transform
// -> P tile in LDS -> P@X (f32 WMMA). Q row-sums via fixed-order LDS
// reduction (deterministic). X tiles are staged LDS-direct by the Tensor
// Data Mover (tensor_load_to_lds, TENSORcnt), double-buffered so the DMA for
// chunk j+1 overlaps the two WMMA passes of chunk j. TDM pad feature
// (4 DWORDs per 128) reproduces the bank-conflict-free 132-float stride.
// ---------------------------------------------------------------------------

typedef float v2f __attribute__((ext_vector_type(2)));
typedef float v8f __attribute__((ext_vector_type(8)));
typedef unsigned v4u __attribute__((ext_vector_type(4)));
typedef int v8i __attribute__((ext_vector_type(8)));

#define NN 4096
#define FD 128
#define GAMMA_INV (1.0f / 6.0f)
#define LAM_F (1.0f / 0.9f - 1.0f)

__device__ __forceinline__ v8f wmma_f32_k4(v2f a, v2f b, v8f c) {
  // (neg_a, A, neg_b, B, c_mod, C, reuse_a, reuse_b)
  return __builtin_amdgcn_wmma_f32_16x16x4_f32(false, a, false, b, (short)0, c,
                                               false, false);
}

// TDM: load a [rows x 128] f32 tile (row stride 128 floats in memory) into
// LDS at byte offset `ldsoff`, inserting 4 pad DWORDs after every 128 DWORDs
// so the LDS row stride is 132 floats. 2D descriptor: groups 0 and 1 only.
__device__ __forceinline__ void tdm_load_tile(const float* gsrc,
                                              unsigned ldsoff, int rows) {
  unsigned long long ga = (unsigned long long)(uintptr_t)gsrc;
  v4u g0;
  g0.x = 1u;                                   // count=1 (valid descriptor)
  g0.y = ldsoff;                               // lds_addr [63:32]
  g0.z = (unsigned)(ga & 0xFFFFFFFFu);         // global_addr lo
  g0.w = (unsigned)((ga >> 32) & 0x1FFFFFFu)   // global_addr hi (57-bit)
         | (2u << 30);                         // type = 2 ("image")
  v8i g1;
  g1[0] = (int)((2u << 16)        // data_size = 4 bytes
                | (1u << 20)      // pad_enable
                | (6u << 22)      // pad_interval: every 128 DWORDs
                | (3u << 25));    // pad_amount: 4 DWORDs
  g1[1] = (int)(128u << 16);      // tensor_dim0[15:0] = 128 (elems)
  g1[2] = (int)(4096u << 16);     // tensor_dim0 hi=0 | tensor_dim1 lo = 4096
  g1[3] = (int)(128u << 16);      // tensor_dim1 hi=0 | tile_dim0 = 128
  g1[4] = rows;                   // tile_dim1 = rows, tile_dim2 = 0
  g1[5] = 128;                    // tensor_dim0_stride = 128 elems
  g1[6] = 0;                      // stride hi / dim1_stride lo
  g1[7] = 0;
  asm volatile("tensor_load_to_lds %0, %1" ::"s"(g0), "s"(g1) : "memory");
}

// --------------------------- degree: D, 1/D, 1/sqrt(D) ----------------------
__global__ __launch_bounds__(256) void degree_kernel(
    const float* __restrict__ A, float* __restrict__ D,
    float* __restrict__ invD, float* __restrict__ invds) {
  __shared__ float red[256];
  const int row = blockIdx.x;
  const float* ar = A + (size_t)row * NN;
  float s = 0.0f;
  for (int j = threadIdx.x; j < NN; j += 256) s += ar[j];
  red[threadIdx.x] = s;
  __syncthreads();
  for (int off = 128; off > 0; off >>= 1) {
    if (threadIdx.x < off) red[threadIdx.x] += red[threadIdx.x + off];
    __syncthreads();
  }
  if (threadIdx.x == 0) {
    float d = red[0] + 1.0f;  // self loop
    D[row] = d;
    invD[row] = 1.0f / d;
    invds[row] = rsqrtf(d);
  }
}

// ------------------- generic f32-WMMA GEMM: C = act(A@B + bias) -------------
__global__ __launch_bounds__(256) void gemm_kernel(
    const float* __restrict__ Ag, const float* __restrict__ Bg,
    const float* __restrict__ bias, float* __restrict__ Cg,
    int M, int Nn, int K, int relu) {
  __shared__ float As[64][34];
  __shared__ float Bs[32][34];
  const int tid = threadIdx.x;
  const int lane = tid & 31;
  const int ln = lane & 15;
  const int kh = lane >> 4;
  const int wv = tid >> 5;
  const int wm = wv >> 1;
  const int wn = wv & 1;
  const int mbase = blockIdx.y * 64;
  const int nbase = blockIdx.x * 32;

  v8f acc = {};
  for (int kb = 0; kb < K; kb += 32) {
    __syncthreads();
    for (int e = tid; e < 64 * 32; e += 256) {
      int r = e >> 5, c = e & 31;
      As[r][c] = Ag[(size_t)(mbase + r) * K + kb + c];
    }
    for (int e = tid; e < 32 * 32; e += 256) {
      int r = e >> 5, c = e & 31;
      Bs[r][c] = Bg[(size_t)(kb + r) * Nn + nbase + c];
    }
    __syncthreads();
#pragma unroll
    for (int kk = 0; kk < 32; kk += 4) {
      const int k0 = kk + 2 * kh;
      v2f a = *(const v2f*)&As[16 * wm + ln][k0];
      v2f b = {Bs[k0][16 * wn + ln], Bs[k0 + 1][16 * wn + ln]};
      acc = wmma_f32_k4(a, b, acc);
    }
  }
#pragma unroll
  for (int r = 0; r < 8; r++) {
    int row = mbase + 16 * wm + r + 8 * kh;
    int col = nbase + 16 * wn + ln;
    float v = acc[r] + bias[col];
    if (relu) v = fmaxf(v, 0.0f);
    Cg[(size_t)row * Nn + col] = v;
  }
}

// ---------------------- prep: X = Fc / sqrt(D), sq = |X|^2 ------------------
__global__ __launch_bounds__(128) void prep_kernel(
    const float* __restrict__ Fc, const float* __restrict__ invds,
    float* __restrict__ X, float* __restrict__ sq) {
  __shared__ float red[128];
  const int row = blockIdx.x;
  const float is = invds[row];
  const float v = Fc[(size_t)row * FD + threadIdx.x] * is;
  X[(size_t)row * FD + threadIdx.x] = v;
  red[threadIdx.x] = v * v;
  __syncthreads();
  for (int off = 64; off > 0; off >>= 1) {
    if (threadIdx.x < off) red[threadIdx.x] += red[threadIdx.x + off];
    __syncthreads();
  }
  if (threadIdx.x == 0) sq[row] = red[0];
}

// ------------------------------- fused prop ---------------------------------
__global__ __launch_bounds__(256) void prop_kernel(
    const float* __restrict__ X, const float* __restrict__ sq,
    const float* __restrict__ A, const float* __restrict__ invD,
    const float* __restrict__ invds, const float* __restrict__ F0,
    float* __restrict__ out) {
  __shared__ float Xi[64][132];     // block's rows of X (TDM pad -> stride 132)
  __shared__ float Xj[2][32][132];  // double-buffered j-chunk rows of X
  __shared__ float Pt[64][34];      // Wm .* A tile
  __shared__ float qsh[64];         // row sums of Wm .* A

  const int tid = threadIdx.x;
  const int lane = tid & 31;
  const int ln = lane & 15;
  const int kh = lane >> 4;
  const int wv = tid >> 5;
  const int wm = wv >> 1;
  const int wn = wv & 1;
  const int ibase = blockIdx.x * 64;

  // Prologue: TDM-stage Xi and the first Xj chunk; LDS written DMA-direct.
  if (wv == 0) {
    tdm_load_tile(X + (size_t)ibase * FD, (unsigned)(uintptr_t)&Xi[0][0], 64);
    tdm_load_tile(X, (unsigned)(uintptr_t)&Xj[0][0][0], 32);
  }
  if (tid < 64) qsh[tid] = 0.0f;

  float sqi[8];
#pragma unroll
  for (int r = 0; r < 8; r++) sqi[r] = sq[ibase + 16 * wm + r + 8 * kh];

  v8f acc0 = {}, acc1 = {}, acc2 = {}, acc3 = {};
  if (wv == 0) __builtin_amdgcn_s_wait_tensorcnt(0);
  __syncthreads();

  for (int jb = 0; jb < NN; jb += 32) {
    const int cur = (jb >> 5) & 1;
    // Kick the DMA for the next chunk; overlaps both WMMA passes below.
    if (wv == 0 && jb + 32 < NN)
      tdm_load_tile(X + (size_t)(jb + 32) * FD,
                    (unsigned)(uintptr_t)&Xj[cur ^ 1][0][0], 32);

    // ---- Gram tile: S(16x16) = Xi[16wm..] @ Xj[16wn..]^T over K=128 ----
    v8f s = {};
#pragma unroll
    for (int kk = 0; kk < FD; kk += 4) {
      const int k0 = kk + 2 * kh;
      v2f a = *(const v2f*)&Xi[16 * wm + ln][k0];
      v2f b = *(const v2f*)&Xj[cur][16 * wn + ln][k0];
      s = wmma_f32_k4(a, b, s);
    }

    // ---- MCP transform: P = max(1 - sqrt(z)/gamma, 0) .* A, diag = 0 ----
    const int jcol = jb + 16 * wn + ln;
    const float sqj = sq[jcol];
#pragma unroll
    for (int r = 0; r < 8; r++) {
      const int mi = 16 * wm + r + 8 * kh;  // local row (C-layout)
      const int gi = ibase + mi;
      float z = fmaxf(sqi[r] + sqj - 2.0f * s[r], 0.0f);
      float w = fmaxf(1.0f - sqrtf(z) * GAMMA_INV, 0.0f);
      if (gi == jcol) w = 0.0f;
      const float aij = A[(size_t)gi * NN + jcol];
      Pt[mi][16 * wn + ln] = w * aij;
    }
    // warm next chunk's adjacency tile in WGP$/L0
    if (jb + 32 < NN)
      __builtin_prefetch(&A[(size_t)(ibase + 16 * wm + 8 * kh) * NN + jcol + 32],
                         0, 3);
    __syncthreads();

    // deterministic Q row-sum (fixed order, no float atomics)
    if (tid < 64) {
      float qs = 0.0f;
#pragma unroll
      for (int c = 0; c < 32; c++) qs += Pt[tid][c];
      qsh[tid] += qs;
    }

    // ---- acc += P(64x32, rows 16wm..) @ Xj(32x128, cols 64wn..) ----
#pragma unroll
    for (int kk = 0; kk < 32; kk += 4) {
      const int k0 = kk + 2 * kh;
      v2f a = *(const v2f*)&Pt[16 * wm + ln][k0];
      const int cb = 64 * wn;
      v2f b0 = {Xj[cur][k0][cb + ln], Xj[cur][k0 + 1][cb + ln]};
      v2f b1 = {Xj[cur][k0][cb + 16 + ln], Xj[cur][k0 + 1][cb + 16 + ln]};
      v2f b2 = {Xj[cur][k0][cb + 32 + ln], Xj[cur][k0 + 1][cb + 32 + ln]};
      v2f b3 = {Xj[cur][k0][cb + 48 + ln], Xj[cur][k0 + 1][cb + 48 + ln]};
      acc0 = wmma_f32_k4(a, b0, acc0);
      acc1 = wmma_f32_k4(a, b1, acc1);
      acc2 = wmma_f32_k4(a, b2, acc2);
      acc3 = wmma_f32_k4(a, b3, acc3);
    }
    // next chunk's DMA must have landed before anyone touches its buffer
    if (wv == 0) __builtin_amdgcn_s_wait_tensorcnt(0);
    __syncthreads();
  }

  // ---- finalize: Fc_next = (invds_i * acc + LAM * F0) / Q ----
#pragma unroll
  for (int r = 0; r < 8; r++) {
    const int mi = 16 * wm + r + 8 * kh;
    const int gi = ibase + mi;
    const float Qi = qsh[mi] * invD[gi] + LAM_F;
    const float rq = 1.0f / Qi;
    const float isd = invds[gi];
    const size_t ro = (size_t)gi * FD;
    const int cb = 64 * wn;
    const int c0 = cb + ln, c1 = cb + 16 + ln, c2 = cb + 32 + ln,
              c3 = cb + 48 + ln;
    out[ro + c0] = (isd * acc0[r] + LAM_F * F0[ro + c0]) * rq;
    out[ro + c1] = (isd * acc1[r] + LAM_F * F0[ro + c1]) * rq;
    out[ro + c2] = (isd * acc2[r] + LAM_F * F0[ro + c2]) * rq;
    out[ro + c3] = (isd * acc3[r] + LAM_F * F0[ro + c3]) * rq;
  }
}

// ------------------------------- launch -------------------------------------
extern "C" void kernel_launch(void* const* d_in, const int* in_sizes, int n_in,
                              void* d_out, int out_size, void* d_ws,
                              size_t ws_size, hipStream_t stream) {
  const float* A  = (const float*)d_in[0];  // [4096,4096]
  const float* F  = (const float*)d_in[1];  // [4096,512]
  const float* W1 = (const float*)d_in[2];  // [512,256]
  const float* b1 = (const float*)d_in[3];  // [256]
  const float* W2 = (const float*)d_in[4];  // [256,128]
  const float* b2 = (const float*)d_in[5];  // [128]
  float* out = (float*)d_out;               // [4096,128]

  float* w = (float*)d_ws;
  float* D     = w;                          // 4096
  float* invD  = D + NN;                     // 4096
  float* invds = invD + NN;                  // 4096
  float* sq    = invds + NN;                 // 4096
  float* F0    = sq + NN;                    // 4096*128
  float* H     = F0 + (size_t)NN * FD;       // 4096*256
  float* X     = H + (size_t)NN * 256;       // 4096*128
  float* FcA   = X + (size_t)NN * FD;        // 4096*128
  float* FcB   = FcA + (size_t)NN * FD;      // 4096*128
  (void)ws_size; (void)in_sizes; (void)n_in; (void)out_size;

  degree_kernel<<<NN, 256, 0, stream>>>(A, D, invD, invds);

  // MLP: H = relu(F@W1+b1); F0 = H@W2+b2
  gemm_kernel<<<dim3(256 / 32, NN / 64), 256, 0, stream>>>(F, W1, b1, H, NN,
                                                           256, 512, 1);
  gemm_kernel<<<dim3(128 / 32, NN / 64), 256, 0, stream>>>(H, W2, b2, F0, NN,
                                                           128, 256, 0);

  const float* cur = F0;
  float* nxt = FcA;
  for (int s = 0; s < 10; s++) {
    prep_kernel<<<NN, 128, 0, stream>>>(cur, invds, X, sq);
    float* dst = (s == 9) ? out : nxt;
    prop_kernel<<<NN / 64, 256, 0, stream>>>(X, sq, A, invD, invds, F0, dst);
    cur = dst;
    nxt = (nxt == FcA) ? FcB : FcA;
  }
}